// AdditiveAttention_11373073400195
// MI455X (gfx1250) — compile-verified
//
#include <hip/hip_runtime.h>
#include <hip/hip_bf16.h>
#include <stdint.h>

typedef __attribute__((ext_vector_type(2))) float v2f;
typedef __attribute__((ext_vector_type(8))) float v8f;

#define BB   4
#define QN   256
#define KN   1024
#define QSZ  256
#define KSZ  256
#define HD   128
#define VD   256

#define QT   4            // q rows per scores block
#define KT   64           // k rows per LDS tile
#define KPAD 4            // LDS row pad (floats) -> conflict-free ds_load_b128
#define NKT  (KN / KT)    // 16 tiles

// ---------------------------------------------------------------------------
// gfx1250 hardware tanh (tanh-insts). Builtin if available, else raw ISA.
// ---------------------------------------------------------------------------
__device__ __forceinline__ float hw_tanh(float x) {
#if __has_builtin(__builtin_amdgcn_tanhf)
    return __builtin_amdgcn_tanhf(x);
#else
    float r;
    asm("v_tanh_f32 %0, %1" : "=v"(r) : "v"(x));
    return r;
#endif
}

// ---------------------------------------------------------------------------
// gfx1250 async global->LDS copy (GLOBAL_LOAD_ASYNC_TO_LDS_B128, ASYNCcnt).
// lds_off = byte offset in LDS (low 32 bits of the flat shared address ARE
// the LDS offset per the aperture rules).
// ---------------------------------------------------------------------------
__device__ __forceinline__ void async_copy_b128(uint32_t lds_off, const float* gaddr) {
    asm volatile("global_load_async_to_lds_b128 %0, %1, off"
                 :: "v"(lds_off), "v"(gaddr)
                 : "memory");
}

template <int N>
__device__ __forceinline__ void wait_async() {
#if __has_builtin(__builtin_amdgcn_s_wait_asynccnt)
    __builtin_amdgcn_s_wait_asynccnt((short)N);
#else
    asm volatile("s_wait_asynccnt %0" :: "i"(N) : "memory");
#endif
}

// ---------------------------------------------------------------------------
// Kernel 1: row-major GEMM  C[M x N] = A[M x Kd] * W[Kd x N]
// One wave per 32x32 C tile: 2 A-frags x 2 B-frags -> 4 WMMAs per k-step.
// f32 WMMA operand layout (ISA 7.12.2):
//   A lane: {A[m][k0], A[m][k0+1]},  m = lane&15, k0 = 2*(lane>>4) (contig)
//   B lane: {W[k0][n], W[k0+1][n]},  n = lane&15                   (stride N)
//   C vgpr r: row m0 + r + 8*(lane>>4), col n0 + (lane&15)
// ---------------------------------------------------------------------------
__global__ __launch_bounds__(32)
void proj_gemm(const float* __restrict__ A, const float* __restrict__ W,
               float* __restrict__ C, int M, int Kd, int N) {
    (void)M;
    const int tiles_n = N >> 5;
    const int m0 = (blockIdx.x / tiles_n) << 5;
    const int n0 = (blockIdx.x % tiles_n) << 5;
    const int lane = threadIdx.x;
    const int half = lane >> 4;
    const int l15  = lane & 15;

    const float* a0 = A + (size_t)(m0 + l15) * Kd + 2 * half;
    const float* a1 = a0 + (size_t)16 * Kd;
    const float* w0 = W + (size_t)(2 * half) * N + n0 + l15;
    const float* w1 = w0 + 16;

    v8f c00 = {}, c01 = {}, c10 = {}, c11 = {};
    for (int k = 0; k < Kd; k += 4) {
        v2f A0 = *(const v2f*)(a0 + k);
        v2f A1 = *(const v2f*)(a1 + k);
        v2f B0, B1;
        B0.x = w0[k * N];  B0.y = w0[k * N + N];
        B1.x = w1[k * N];  B1.y = w1[k * N + N];
        c00 = __builtin_amdgcn_wmma_f32_16x16x4_f32(false, A0, false, B0, (short)0, c00, false, false);
        c01 = __builtin_amdgcn_wmma_f32_16x16x4_f32(false, A0, false, B1, (short)0, c01, false, false);
        c10 = __builtin_amdgcn_wmma_f32_16x16x4_f32(false, A1, false, B0, (short)0, c10, false, false);
        c11 = __builtin_amdgcn_wmma_f32_16x16x4_f32(false, A1, false, B1, (short)0, c11, false, false);
    }

    float* cr0 = C + (size_t)(m0 + 8 * half) * N + n0 + l15;
    float* cr1 = C + (size_t)(m0 + 16 + 8 * half) * N + n0 + l15;
#pragma unroll
    for (int r = 0; r < 8; ++r) {
        cr0[(size_t)r * N]      = c00[r];
        cr0[(size_t)r * N + 16] = c01[r];
        cr1[(size_t)r * N]      = c10[r];
        cr1[(size_t)r * N + 16] = c11[r];
    }
}

// ---------------------------------------------------------------------------
// Kernel 2: fused scores + softmax.
// Block = 256 threads = QT(4) q-rows x KT(64) k-columns.
// kh double-buffered through LDS with async global->LDS b128 copies:
// tile i+1 streams in (ASYNCcnt) while v_tanh_f32 consumes tile i from LDS.
// Per-row softmax: wave32 shfl reduce + 2-wave LDS combine, v_exp_f32.
// ---------------------------------------------------------------------------
__global__ __launch_bounds__(256)
void scores_softmax(const float* __restrict__ qh, const float* __restrict__ kh,
                    const float* __restrict__ wv, float* __restrict__ attn) {
    __shared__ float s_k[2][KT][HD + KPAD];  // 67.6 KB double-buffered kh tile
    __shared__ float s_q[QT][HD];            // 2 KB
    __shared__ float s_w[HD];                // 0.5 KB
    __shared__ float s_sc[QT][KN];           // 16 KB scores
    __shared__ float s_red[16];

    const int t  = threadIdx.x;
    const int b  = blockIdx.x / (QN / QT);
    const int q0 = (blockIdx.x % (QN / QT)) * QT;

    // stage 4 q-rows (512 floats) + w_v
    {
        const int i0 = t, i1 = t + 256;
        s_q[i0 >> 7][i0 & 127] =
            qh[(size_t)(b * QN + q0 + (i0 >> 7)) * HD + (i0 & 127)];
        s_q[i1 >> 7][i1 & 127] =
            qh[(size_t)(b * QN + q0 + (i1 >> 7)) * HD + (i1 & 127)];
        if (t < HD) s_w[t] = wv[t];
    }

    const float* kbase = kh + (size_t)b * KN * HD;
    const int ql = t >> 6;     // 0..3  : q row within block
    const int kl = t & 63;     // 0..63 : k row within tile

    // async-stage one 64x128 tile (2048 float4, coalesced): 8 loads/thread
    auto issue_tile = [&](int tile, int buf) {
        const int k0 = tile * KT;
#pragma unroll
        for (int j = 0; j < 8; ++j) {
            const int f   = t + 256 * j;
            const int row = f >> 5;
            const int col = (f & 31) << 2;
            async_copy_b128((uint32_t)(uintptr_t)&s_k[buf][row][col],
                            kbase + (size_t)(k0 + row) * HD + col);
        }
    };

    issue_tile(0, 0);
    __syncthreads();           // covers the s_q/s_w VALU stores too

    for (int tile = 0; tile < NKT; ++tile) {
        const int cur = tile & 1;
        if (tile + 1 < NKT) {
            issue_tile(tile + 1, cur ^ 1);  // buffer freed by iter(tile-1) end barrier
            wait_async<8>();                // drain tile's 8 loads; keep 8 in flight
        } else {
            wait_async<0>();
        }
        __syncthreads();       // all waves' tile data visible in LDS

        float acc = 0.f;
#pragma unroll
        for (int h = 0; h < HD; h += 4) {
            const float4 k4 = *(const float4*)&s_k[cur][kl][h];
            const float4 q4 = *(const float4*)&s_q[ql][h];
            const float4 w4 = *(const float4*)&s_w[h];
            acc += w4.x * hw_tanh(q4.x + k4.x);
            acc += w4.y * hw_tanh(q4.y + k4.y);
            acc += w4.z * hw_tanh(q4.z + k4.z);
            acc += w4.w * hw_tanh(q4.w + k4.w);
        }
        s_sc[ql][tile * KT + kl] = acc;
        __syncthreads();       // everyone done reading s_k[cur] before reuse
    }

    // ---- softmax over row ql (this thread owns k = kl + 64*i) ----
    float vals[NKT];
    float mx = -3.402823466e38f;
#pragma unroll
    for (int i = 0; i < NKT; ++i) {
        vals[i] = s_sc[ql][kl + KT * i];   // own writes: in-order, no barrier
        mx = fmaxf(mx, vals[i]);
    }
#pragma unroll
    for (int off = 16; off > 0; off >>= 1)
        mx = fmaxf(mx, __shfl_xor(mx, off, 32));
    const int wid = t >> 5;                // ql == wid >> 1
    if ((t & 31) == 0) s_red[wid] = mx;
    __syncthreads();
    mx = fmaxf(s_red[2 * ql], s_red[2 * ql + 1]);

    float e[NKT];
    float sum = 0.f;
#pragma unroll
    for (int i = 0; i < NKT; ++i) { e[i] = __expf(vals[i] - mx); sum += e[i]; }
#pragma unroll
    for (int off = 16; off > 0; off >>= 1)
        sum += __shfl_xor(sum, off, 32);
    if ((t & 31) == 0) s_red[8 + wid] = sum;
    __syncthreads();
    sum = s_red[8 + 2 * ql] + s_red[8 + 2 * ql + 1];

    const float inv = 1.0f / sum;
    float* arow = attn + (size_t)(b * QN + q0 + ql) * KN;
#pragma unroll
    for (int i = 0; i < NKT; ++i) arow[kl + KT * i] = e[i] * inv;
}

// ---------------------------------------------------------------------------
// Kernel 3: out[b,q,d] = sum_k attn[b,q,k] * values[b,d,k]
// One wave per 32x32 tile; BOTH WMMA operands are contiguous float2 per lane
// (values is (D,K) so B[k][n]=V[n][k] pairs are adjacent in memory).
// ---------------------------------------------------------------------------
__global__ __launch_bounds__(32)
void attn_v_gemm(const float* __restrict__ attn, const float* __restrict__ vals,
                 float* __restrict__ out) {
    const int tiles_n = VD >> 5;                 // 8
    const int tiles_m = QN >> 5;                 // 8
    const int b  = blockIdx.x / (tiles_m * tiles_n);
    const int tt = blockIdx.x % (tiles_m * tiles_n);
    const int m0 = (tt / tiles_n) << 5;
    const int n0 = (tt % tiles_n) << 5;

    const int lane = threadIdx.x;
    const int half = lane >> 4;
    const int l15  = lane & 15;

    const float* a0 = attn + (size_t)b * QN * KN + (size_t)(m0 + l15) * KN + 2 * half;
    const float* a1 = a0 + (size_t)16 * KN;
    const float* b0 = vals + (size_t)b * VD * KN + (size_t)(n0 + l15) * KN + 2 * half;
    const float* b1 = b0 + (size_t)16 * KN;

    v8f c00 = {}, c01 = {}, c10 = {}, c11 = {};
    for (int k = 0; k < KN; k += 4) {
        v2f A0 = *(const v2f*)(a0 + k);
        v2f A1 = *(const v2f*)(a1 + k);
        v2f B0 = *(const v2f*)(b0 + k);
        v2f B1 = *(const v2f*)(b1 + k);
        c00 = __builtin_amdgcn_wmma_f32_16x16x4_f32(false, A0, false, B0, (short)0, c00, false, false);
        c01 = __builtin_amdgcn_wmma_f32_16x16x4_f32(false, A0, false, B1, (short)0, c01, false, false);
        c10 = __builtin_amdgcn_wmma_f32_16x16x4_f32(false, A1, false, B0, (short)0, c10, false, false);
        c11 = __builtin_amdgcn_wmma_f32_16x16x4_f32(false, A1, false, B1, (short)0, c11, false, false);
    }

    float* cr0 = out + (size_t)b * QN * VD + (size_t)(m0 + 8 * half) * VD + n0 + l15;
    float* cr1 = cr0 + (size_t)16 * VD;
#pragma unroll
    for (int r = 0; r < 8; ++r) {
        cr0[(size_t)r * VD]      = c00[r];
        cr0[(size_t)r * VD + 16] = c01[r];
        cr1[(size_t)r * VD]      = c10[r];
        cr1[(size_t)r * VD + 16] = c11[r];
    }
}

// ---------------------------------------------------------------------------
extern "C" void kernel_launch(void* const* d_in, const int* in_sizes, int n_in,
                              void* d_out, int out_size, void* d_ws, size_t ws_size,
                              hipStream_t stream) {
    (void)in_sizes; (void)n_in; (void)out_size; (void)ws_size;

    const float* queries = (const float*)d_in[0];   // (B,Q,256)
    const float* keys    = (const float*)d_in[1];   // (B,K,256)
    const float* values  = (const float*)d_in[2];   // (B,VD,K)
    const float* W_q     = (const float*)d_in[3];   // (256,128)
    const float* W_k     = (const float*)d_in[4];   // (256,128)
    const float* w_v     = (const float*)d_in[5];   // (128,)
    float*       out     = (float*)d_out;           // (B,Q,VD)

    float* qh   = (float*)d_ws;                     // B*QN*HD   = 131072 f
    float* kh   = qh + (size_t)BB * QN * HD;        // B*KN*HD   = 524288 f
    float* attn = kh + (size_t)BB * KN * HD;        // B*QN*KN   = 1048576 f

    // 1) projections (WMMA f32, 32x32 tiles/wave)
    proj_gemm<<<(BB * QN / 32) * (HD / 32), 32, 0, stream>>>(
        queries, W_q, qh, BB * QN, QSZ, HD);
    proj_gemm<<<(BB * KN / 32) * (HD / 32), 32, 0, stream>>>(
        keys, W_k, kh, BB * KN, KSZ, HD);

    // 2) fused tanh-scores + softmax (double-buffered async LDS staging)
    scores_softmax<<<BB * QN / QT, 256, 0, stream>>>(qh, kh, w_v, attn);

    // 3) attn x V^T (WMMA f32, 32x32 tiles/wave)
    attn_v_gemm<<<BB * (QN / 32) * (VD / 32), 32, 0, stream>>>(attn, values, out);
}